// ServiceGCN_67224828117292
// MI455X (gfx1250) — compile-verified
//
#include <hip/hip_runtime.h>
#include <hip/hip_bf16.h>

typedef __attribute__((ext_vector_type(2))) float v2f;
typedef __attribute__((ext_vector_type(8))) float v8f;

#define BN_EPS 1e-5f

// ---------------- degree / normalization ----------------
__global__ void deg_init_kernel(float* __restrict__ deg, int n) {
  int i = blockIdx.x * blockDim.x + threadIdx.x;
  if (i < n) deg[i] = 1.0f;  // self-loop contributes 1 to every node's degree
}

__global__ void deg_edge_kernel(const int* __restrict__ dst, float* __restrict__ deg, int e) {
  int i = blockIdx.x * blockDim.x + threadIdx.x;
  if (i < e) unsafeAtomicAdd(&deg[dst[i]], 1.0f);
}

__global__ void dinv_kernel(float* __restrict__ deg, int n) {
  int i = blockIdx.x * blockDim.x + threadIdx.x;
  if (i < n) {
    float d = deg[i];
    deg[i] = (d > 0.0f) ? rsqrtf(d) : 0.0f;  // in-place: deg -> deg^{-1/2}
  }
}

// ---------------- dense GEMM via V_WMMA_F32_16X16X4_F32 ----------------
// One wave32 computes a 16x16 fp32 output tile; K accumulated in steps of 4.
// KDIM and NCOLS are compile-time so every global load folds to an
// immediate-offset address (no per-iteration 64-bit VALU address math).
// A (16x4) layout: lanes 0-15 hold M=lane, VGPR0/1 = K0,K1; lanes 16-31 = K2,K3.
// B (4x16) layout: lanes hold N; VGPR0/1 = K0,K1 (lanes 0-15) / K2,K3 (lanes 16-31).
template <int KDIM, int NCOLS>
__global__ void gemm_wmma_f32(const float* __restrict__ A, const float* __restrict__ B,
                              float* __restrict__ C, int Mrows) {
  const int wave = threadIdx.x >> 5;
  const int lane = threadIdx.x & 31;
  constexpr int tilesPerRow = NCOLS / 16;
  const int numTiles = (Mrows >> 4) * tilesPerRow;
  const int tile = blockIdx.x * (blockDim.x >> 5) + wave;
  if (tile >= numTiles) return;  // wave-uniform: EXEC stays all-1s for WMMA

  const int trow = tile / tilesPerRow;
  const int tcol = tile - trow * tilesPerRow;
  const int half = lane >> 4;   // 0: K pair {0,1}, 1: K pair {2,3}
  const int l    = lane & 15;

  const float* arow = A + (size_t)(trow * 16 + l) * KDIM + half * 2;
  const float* bcol = B + (size_t)half * 2 * NCOLS + (tcol * 16 + l);

  // Preload the full B panel for this tile column: KDIM/4 WMMA steps x 2 floats.
  v2f breg[KDIM / 4];
#pragma unroll
  for (int i = 0; i < KDIM / 4; ++i) {
    breg[i].x = bcol[(size_t)(i * 4 + 0) * NCOLS];   // constexpr immediate offsets
    breg[i].y = bcol[(size_t)(i * 4 + 1) * NCOLS];
  }

  v8f acc = {0.f, 0.f, 0.f, 0.f, 0.f, 0.f, 0.f, 0.f};
#pragma unroll
  for (int i = 0; i < KDIM / 4; ++i) {
    v2f a = *reinterpret_cast<const v2f*>(arow + i * 4);  // global_load_b64, imm offset
    acc = __builtin_amdgcn_wmma_f32_16x16x4_f32(false, a, false, breg[i], (short)0,
                                                acc, false, false);
  }

  // C/D layout: VGPR r -> row r (lanes 0-15) / row r+8 (lanes 16-31)
  float* crow = C + (size_t)(trow * 16 + half * 8) * NCOLS + (tcol * 16 + l);
#pragma unroll
  for (int r = 0; r < 8; ++r) crow[(size_t)r * NCOLS] = acc[r];
}

// ---------------- aggregation ----------------
// agg[i,c] = bias[c] + h[i,c] * dinv[i]^2   (bias + self-loop message)
__global__ void agg_init_kernel(const float* __restrict__ h, const float* __restrict__ dinv,
                                const float* __restrict__ bias, float* __restrict__ agg,
                                long long total, int log2C) {
  long long idx = (long long)blockIdx.x * blockDim.x + threadIdx.x;
  if (idx >= total) return;
  int C = 1 << log2C;
  int i = (int)(idx >> log2C);
  int c = (int)(idx & (C - 1));
  float di = dinv[i];
  agg[idx] = bias[c] + h[idx] * di * di;
}

// agg[dst,c] += h[src,c] * dinv[src] * dinv[dst], float4 per thread
__global__ void scatter_edge_kernel(const float* __restrict__ h, const int* __restrict__ src,
                                    const int* __restrict__ dst, const float* __restrict__ dinv,
                                    float* __restrict__ agg, long long total, int C,
                                    int log2Chunks) {
  long long tid = (long long)blockIdx.x * blockDim.x + threadIdx.x;
  if (tid >= total) return;
  const int chunks = 1 << log2Chunks;
  const int e  = (int)(tid >> log2Chunks);
  const int ch = (int)(tid & (chunks - 1));
  const int s = src[e];
  const int d = dst[e];
  const float nrm = dinv[s] * dinv[d];
  const float4 v = *reinterpret_cast<const float4*>(h + (size_t)s * C + ch * 4);
  float* base = agg + (size_t)d * C + ch * 4;
  unsafeAtomicAdd(base + 0, v.x * nrm);
  unsafeAtomicAdd(base + 1, v.y * nrm);
  unsafeAtomicAdd(base + 2, v.z * nrm);
  unsafeAtomicAdd(base + 3, v.w * nrm);
}

// ---------------- batch norm ----------------
__global__ void zero_kernel(float* __restrict__ p, int n) {
  int i = blockIdx.x * blockDim.x + threadIdx.x;
  if (i < n) p[i] = 0.0f;
}

// blockDim.x == C; each block reduces a row range, then a few float atomics
__global__ void bn_stats_kernel(const float* __restrict__ h, float* __restrict__ sums,
                                float* __restrict__ sumsq, int Nrows, int C, int rowsPerBlock) {
  int c  = threadIdx.x;
  int r0 = blockIdx.x * rowsPerBlock;
  int r1 = min(r0 + rowsPerBlock, Nrows);
  float s = 0.0f, q = 0.0f;
  for (int r = r0; r < r1; ++r) {
    float v = h[(size_t)r * C + c];
    s += v;
    q += v * v;
  }
  unsafeAtomicAdd(&sums[c], s);
  unsafeAtomicAdd(&sumsq[c], q);
}

__global__ void bn_finalize_kernel(const float* __restrict__ sums, const float* __restrict__ sumsq,
                                   const float* __restrict__ gamma, const float* __restrict__ beta,
                                   float* __restrict__ scale, float* __restrict__ shift,
                                   int Nrows, int C) {
  int c = blockIdx.x * blockDim.x + threadIdx.x;
  if (c >= C) return;
  float inv_n = 1.0f / (float)Nrows;
  float m   = sums[c] * inv_n;
  float var = sumsq[c] * inv_n - m * m;
  float sc  = rsqrtf(var + BN_EPS) * gamma[c];
  scale[c] = sc;
  shift[c] = beta[c] - m * sc;
}

// out = relu(h * scale + shift)
__global__ void bn_apply_relu_kernel(const float* __restrict__ h, const float* __restrict__ scale,
                                     const float* __restrict__ shift, float* __restrict__ out,
                                     long long total, int log2C) {
  long long idx = (long long)blockIdx.x * blockDim.x + threadIdx.x;
  if (idx >= total) return;
  int c = (int)(idx & ((1 << log2C) - 1));
  float v = h[idx] * scale[c] + shift[c];
  out[idx] = fmaxf(v, 0.0f);
}

// mean over rows of batch-normalized output; per-column mean of normalized data
// is exactly m*scale + shift (== beta analytically)
__global__ void bn_mean_out_kernel(const float* __restrict__ sums, const float* __restrict__ sumsq,
                                   const float* __restrict__ gamma, const float* __restrict__ beta,
                                   float* __restrict__ out, int Nrows, int C) {
  int c = blockIdx.x * blockDim.x + threadIdx.x;
  if (c >= C) return;
  float inv_n = 1.0f / (float)Nrows;
  float m   = sums[c] * inv_n;
  float var = sumsq[c] * inv_n - m * m;
  float sc  = rsqrtf(var + BN_EPS) * gamma[c];
  float sh  = beta[c] - m * sc;
  out[c] = m * sc + sh;
}

// ---------------- host ----------------
extern "C" void kernel_launch(void* const* d_in, const int* in_sizes, int n_in,
                              void* d_out, int out_size, void* d_ws, size_t ws_size,
                              hipStream_t stream) {
  const float* x      = (const float*)d_in[0];
  const int*   ei     = (const int*)d_in[1];
  const float* W1     = (const float*)d_in[2];
  const float* b1     = (const float*)d_in[3];
  const float* gamma1 = (const float*)d_in[4];
  const float* beta1  = (const float*)d_in[5];
  const float* W2     = (const float*)d_in[6];
  const float* b2     = (const float*)d_in[7];
  const float* gamma2 = (const float*)d_in[8];
  const float* beta2  = (const float*)d_in[9];
  float* out = (float*)d_out;

  const int Chid = in_sizes[3];          // 64
  const int Cin  = in_sizes[2] / Chid;   // 64
  const int Cout = in_sizes[7];          // 128
  const int N    = in_sizes[0] / Cin;    // 100000
  const int E    = in_sizes[1] / 2;      // 3200000
  const int* src = ei;
  const int* dst = ei + E;

  const int log2Chid = 31 - __builtin_clz(Chid);
  const int log2Cout = 31 - __builtin_clz(Cout);

  // workspace layout (floats)
  float* w = (float*)d_ws;
  float* dinv  = w;  w += N;
  float* bufA  = w;  w += (size_t)N * Chid;   // H1, then reused as relu(BN(agg1))
  float* bufB  = w;  w += (size_t)N * Chid;   // agg1
  float* buf2A = w;  w += (size_t)N * Cout;   // H2 = relu_out @ W2
  float* buf2B = w;  w += (size_t)N * Cout;   // agg2
  float* sums  = w;  w += 128;
  float* sumsq = w;  w += 128;
  float* scale = w;  w += 128;
  float* shift = w;  w += 128;

  const int TPB = 256;
  const int ROWS_PER_BLOCK = 1024;
  const int statBlocks = (N + ROWS_PER_BLOCK - 1) / ROWS_PER_BLOCK;

  // degrees -> dinv
  deg_init_kernel<<<(N + TPB - 1) / TPB, TPB, 0, stream>>>(dinv, N);
  deg_edge_kernel<<<(E + TPB - 1) / TPB, TPB, 0, stream>>>(dst, dinv, E);
  dinv_kernel<<<(N + TPB - 1) / TPB, TPB, 0, stream>>>(dinv, N);

  // ---- layer 1: h1 = x @ W1 (fp32 WMMA) ----
  {
    int tiles = (N / 16) * (Chid / 16);
    gemm_wmma_f32<64, 64><<<(tiles + 7) / 8, TPB, 0, stream>>>(x, W1, bufA, N);
  }
  {
    long long total = (long long)N * Chid;
    agg_init_kernel<<<(unsigned)((total + TPB - 1) / TPB), TPB, 0, stream>>>(
        bufA, dinv, b1, bufB, total, log2Chid);
    long long etotal = (long long)E * (Chid / 4);
    scatter_edge_kernel<<<(unsigned)((etotal + TPB - 1) / TPB), TPB, 0, stream>>>(
        bufA, src, dst, dinv, bufB, etotal, Chid, log2Chid - 2);
  }
  zero_kernel<<<1, 256, 0, stream>>>(sums, 256);  // sums + sumsq contiguous
  bn_stats_kernel<<<statBlocks, Chid, 0, stream>>>(bufB, sums, sumsq, N, Chid, ROWS_PER_BLOCK);
  bn_finalize_kernel<<<1, 128, 0, stream>>>(sums, sumsq, gamma1, beta1, scale, shift, N, Chid);
  {
    long long total = (long long)N * Chid;
    bn_apply_relu_kernel<<<(unsigned)((total + TPB - 1) / TPB), TPB, 0, stream>>>(
        bufB, scale, shift, bufA, total, log2Chid);
  }

  // ---- layer 2: h2 = relu_out @ W2 (fp32 WMMA) ----
  {
    int tiles = (N / 16) * (Cout / 16);
    gemm_wmma_f32<64, 128><<<(tiles + 7) / 8, TPB, 0, stream>>>(bufA, W2, buf2A, N);
  }
  {
    long long total = (long long)N * Cout;
    agg_init_kernel<<<(unsigned)((total + TPB - 1) / TPB), TPB, 0, stream>>>(
        buf2A, dinv, b2, buf2B, total, log2Cout);
    long long etotal = (long long)E * (Cout / 4);
    scatter_edge_kernel<<<(unsigned)((etotal + TPB - 1) / TPB), TPB, 0, stream>>>(
        buf2A, src, dst, dinv, buf2B, etotal, Cout, log2Cout - 2);
  }
  zero_kernel<<<1, 256, 0, stream>>>(sums, 256);
  bn_stats_kernel<<<statBlocks, Cout, 0, stream>>>(buf2B, sums, sumsq, N, Cout, ROWS_PER_BLOCK);
  bn_mean_out_kernel<<<1, 128, 0, stream>>>(sums, sumsq, gamma2, beta2, out, N, Cout);
}